// WeightedHighFrequencyLoss_74801150427535
// MI455X (gfx1250) — compile-verified
//
#include <hip/hip_runtime.h>
#include <hip/hip_bf16.h>

// ---------------------------------------------------------------------------
// WeightedHighFrequencyLoss for MI455X (gfx1250), bandwidth-bound.
// pred/gt: (8,3,1024,1024) fp32. out: (24,1,1024,1024) fp32.
// Pipeline: haar-diff (async-LDS staged, double buffered) -> min/max over
// upsampled field -> upsample+normalize with NT stores.
// ---------------------------------------------------------------------------

typedef float f32x4 __attribute__((ext_vector_type(4)));
typedef float f32x2 __attribute__((ext_vector_type(2)));

static constexpr int IMGS = 24;        // B*C
static constexpr int HW   = 1024;      // full-res H = W
static constexpr int HWH  = 512;       // half-res
static constexpr int ROWS_PER_BLOCK = 8;

// ---- CDNA5 async global->LDS copy (16 bytes per lane) ----------------------
// VDST = VGPR holding the LDS byte address. Per the flat-aperture rule
// the low 32 bits of a generic pointer to
// __shared__ memory are exactly the LDS byte address, so no addrspace casts
// are needed. VADDR = 64-bit global VA. Tracked by ASYNCcnt.
__device__ __forceinline__ void async_copy16(const float* g, float* lds_generic) {
  unsigned lds_addr = (unsigned)(unsigned long long)(void*)lds_generic;
  asm volatile("global_load_async_to_lds_b128 %0, %1, off"
               :: "v"(lds_addr), "v"(g) : "memory");
}

__device__ __forceinline__ void wait_async_0() {
  asm volatile("s_wait_asynccnt 0" ::: "memory");
}
__device__ __forceinline__ void wait_async_4() {
  asm volatile("s_wait_asynccnt 4" ::: "memory");
}

// ---- Haar high-frequency magnitude of one 2x2 block ------------------------
__device__ __forceinline__ float haar_hf(float a, float b, float c, float d) {
  float t1 = a + b - c - d;   // 2*cH
  float t2 = a - b + c - d;   // 2*cV
  float t3 = a - b - c + d;   // 2*cD
  return 0.5f * sqrtf(t1 * t1 + t2 * t2 + t3 * t3);
}

// ---- kernel 0: init global min/max cells -----------------------------------
__global__ void init_mm(unsigned* mm) {
  mm[0] = 0x7F7FFFFFu;  // bits of FLT_MAX (all wm values are >= 0)
  mm[1] = 0x00000000u;  // bits of 0.0f
}

// ---- kernel 1: |HF(pred)-HF(gt)| at half resolution ------------------------
// One block covers ROWS_PER_BLOCK half-rows; inputs staged via async-to-LDS
// with double buffering (32 KB LDS). Each lane reads back only its own slots,
// so only s_wait_asynccnt (per-wave) is needed, no barrier.
__global__ void __launch_bounds__(256)
haar_diff(const float* __restrict__ pred, const float* __restrict__ gt,
          float* __restrict__ wmh) {
  __shared__ float stage[2][4][HW];  // [buf][pred0,pred1,gt0,gt1][1024] = 32KB
  const int t  = threadIdx.x;
  const int c4 = t * 4;
  const int base_hr = blockIdx.x * ROWS_PER_BLOCK;

  auto issue = [&](int r) {
    const int hr  = base_hr + r;
    const int img = hr >> 9;
    const int hy  = hr & (HWH - 1);
    const size_t row0 = (size_t)img * (HW * HW) + (size_t)(hy * 2) * HW;
    const int b = r & 1;
    async_copy16(pred + row0 + c4,      &stage[b][0][c4]);
    async_copy16(pred + row0 + HW + c4, &stage[b][1][c4]);
    async_copy16(gt   + row0 + c4,      &stage[b][2][c4]);
    async_copy16(gt   + row0 + HW + c4, &stage[b][3][c4]);
  };

  issue(0);
  for (int r = 0; r < ROWS_PER_BLOCK; ++r) {
    if (r + 1 < ROWS_PER_BLOCK) { issue(r + 1); wait_async_4(); }
    else                        { wait_async_0(); }
    const int b = r & 1;
    const f32x4 p0 = *(const f32x4*)&stage[b][0][c4];
    const f32x4 p1 = *(const f32x4*)&stage[b][1][c4];
    const f32x4 g0 = *(const f32x4*)&stage[b][2][c4];
    const f32x4 g1 = *(const f32x4*)&stage[b][3][c4];
    f32x2 o;
    o.x = fabsf(haar_hf(p0.x, p0.y, p1.x, p1.y) - haar_hf(g0.x, g0.y, g1.x, g1.y));
    o.y = fabsf(haar_hf(p0.z, p0.w, p1.z, p1.w) - haar_hf(g0.z, g0.w, g1.z, g1.w));
    const int hr  = base_hr + r;
    const int img = hr >> 9;
    const int hy  = hr & (HWH - 1);
    *(f32x2*)&wmh[(size_t)img * (HWH * HWH) + (size_t)hy * HWH + t * 2] = o;
  }
}

// ---- bilinear helpers (jax.image.resize half-pixel convention) -------------
// src coordinate for output index i at scale 2x: s = 0.5*i - 0.25
__device__ __forceinline__ float lerp_row(const float* r, int x) {
  const int k   = x >> 1;
  const int odd = x & 1;
  const int i0  = odd ? k : (k - 1 < 0 ? 0 : k - 1);
  const int i1  = odd ? (k + 1 > HWH - 1 ? HWH - 1 : k + 1) : k;
  const float w1 = odd ? 0.25f : 0.75f;
  const float a = r[i0];
  return a + (r[i1] - a) * w1;
}

__device__ __forceinline__ void block_reduce_minmax(float lmin, float lmax,
                                                    unsigned* mm) {
  for (int off = 16; off > 0; off >>= 1) {
    lmin = fminf(lmin, __shfl_xor(lmin, off, 32));
    lmax = fmaxf(lmax, __shfl_xor(lmax, off, 32));
  }
  __shared__ float smin[8], smax[8];
  const int lane = threadIdx.x & 31, wid = threadIdx.x >> 5;
  if (lane == 0) { smin[wid] = lmin; smax[wid] = lmax; }
  __syncthreads();
  if (threadIdx.x == 0) {
    float m = smin[0], M = smax[0];
    #pragma unroll
    for (int i = 1; i < 8; ++i) { m = fminf(m, smin[i]); M = fmaxf(M, smax[i]); }
    atomicMin(&mm[0], __float_as_uint(m));   // valid: all values >= 0
    atomicMax(&mm[1], __float_as_uint(M));
  }
}

// ---- kernels 2/3: upsample (reduce-only, then normalize+store) -------------
// One block = one output row. The two contributing half-rows are staged into
// LDS with async loads (cross-wave sharing -> per-wave wait, then barrier).
// Half-res field (25 MB) is L2-resident, so the double pass costs almost no
// HBM traffic.
template <bool WRITE>
__global__ void __launch_bounds__(256)
upsample_pass(const float* __restrict__ wmh, unsigned* __restrict__ mm,
              float* __restrict__ out) {
  __shared__ float r0[HWH], r1[HWH];
  const int t   = threadIdx.x;
  const int row = blockIdx.x;             // img*1024 + y
  const int img = row >> 10;
  const int y   = row & (HW - 1);

  const float sy = 0.5f * (float)y - 0.25f;
  const float fy = sy - floorf(sy);       // 0.25 or 0.75
  int iy0 = (int)floorf(sy);
  int iy1 = iy0 + 1;
  iy0 = iy0 < 0 ? 0 : (iy0 > HWH - 1 ? HWH - 1 : iy0);
  iy1 = iy1 < 0 ? 0 : (iy1 > HWH - 1 ? HWH - 1 : iy1);

  const float* src0 = wmh + (size_t)img * (HWH * HWH) + (size_t)iy0 * HWH;
  const float* src1 = wmh + (size_t)img * (HWH * HWH) + (size_t)iy1 * HWH;
  if (t < 128) async_copy16(src0 + t * 4,         &r0[t * 4]);
  else         async_copy16(src1 + (t - 128) * 4, &r1[(t - 128) * 4]);
  wait_async_0();
  __syncthreads();

  float res[4];
  #pragma unroll
  for (int j = 0; j < 4; ++j) {
    const int x = t * 4 + j;
    const float v0 = lerp_row(r0, x);
    const float v1 = lerp_row(r1, x);
    res[j] = v0 + fy * (v1 - v0);
  }

  if (!WRITE) {
    const float lmin = fminf(fminf(res[0], res[1]), fminf(res[2], res[3]));
    const float lmax = fmaxf(fmaxf(res[0], res[1]), fmaxf(res[2], res[3]));
    block_reduce_minmax(lmin, lmax, mm);
  } else {
    const float mn  = __uint_as_float(mm[0]);
    const float mx  = __uint_as_float(mm[1]);
    const float inv = 1.0f / (mx - mn + 1e-8f);
    f32x4 o;
    o.x = (res[0] - mn) * inv;
    o.y = (res[1] - mn) * inv;
    o.z = (res[2] - mn) * inv;
    o.w = (res[3] - mn) * inv;
    // write-once output: non-temporal so it doesn't evict the L2-resident
    // half-res field.
    __builtin_nontemporal_store(o, (f32x4*)(out + (size_t)row * HW + t * 4));
  }
}

// ---------------------------------------------------------------------------
extern "C" void kernel_launch(void* const* d_in, const int* in_sizes, int n_in,
                              void* d_out, int out_size, void* d_ws, size_t ws_size,
                              hipStream_t stream) {
  const float* pred = (const float*)d_in[0];
  const float* gt   = (const float*)d_in[1];
  float* out        = (float*)d_out;

  unsigned* mm = (unsigned*)d_ws;                       // [min_bits, max_bits]
  float* wmh   = (float*)((char*)d_ws + 256);           // 24*512*512 fp32 (25 MB)

  init_mm<<<1, 1, 0, stream>>>(mm);

  const int total_half_rows = IMGS * HWH;               // 12288
  haar_diff<<<total_half_rows / ROWS_PER_BLOCK, 256, 0, stream>>>(pred, gt, wmh);

  const int total_rows = IMGS * HW;                     // 24576
  upsample_pass<false><<<total_rows, 256, 0, stream>>>(wmh, mm, nullptr);
  upsample_pass<true ><<<total_rows, 256, 0, stream>>>(wmh, mm, out);
}